// Model_43224550868016
// MI455X (gfx1250) — compile-verified
//
#include <hip/hip_runtime.h>

// ---------------------------------------------------------------------------
// 2-layer GCN for MI455X (gfx1250).
// Roofline: scatter/gather of edge messages (~2.6 GB random traffic) dominates;
// GEMMs are ~10 GFLOP -> use full-precision V_WMMA_F32_16X16X4_F32 so the
// matrix path is exact fp32 while still exercising the CDNA5 WMMA pipe.
// ---------------------------------------------------------------------------

typedef __attribute__((ext_vector_type(2))) float v2f;
typedef __attribute__((ext_vector_type(8))) float v8f;

// ----------------------------- degree prep ---------------------------------
__global__ void k_init_deg(float* __restrict__ deg, int n) {
  int i = blockIdx.x * blockDim.x + threadIdx.x;
  if (i < n) deg[i] = 1.0f;  // self-loop contributes 1 to every node's degree
}

__global__ void k_count_deg(const int* __restrict__ dst, float* __restrict__ deg, int e) {
  int i = blockIdx.x * blockDim.x + threadIdx.x;
  if (i < e) unsafeAtomicAdd(&deg[dst[i]], 1.0f);  // global_atomic_add_f32, no-return
}

__global__ void k_deg_to_dinv(float* __restrict__ deg, int n) {
  int i = blockIdx.x * blockDim.x + threadIdx.x;
  if (i < n) deg[i] = rsqrtf(deg[i]);  // deg >= 1 always
}

// ------------------------- fp32 WMMA GEMM ----------------------------------
// C[M x N] = A[M x K] @ B[K x N], all row-major fp32.
// One wave computes one 16x16 tile of C; K consumed 4 at a time by
// V_WMMA_F32_16X16X4_F32. Requires M%16==0, N%16==0, K%4==0 (true here:
// M=50000=3125*16, N in {256,128}, K in {256}).
//
// VGPR layouts (ISA 7.12.2):
//   A 16x4 : lane m = lane%16;  v0/v1 hold K = (lane/16)*2 + {0,1}
//   B 4x16 : lane n = lane%16;  v0/v1 hold K = (lane/16)*2 + {0,1}
//   C 16x16: VGPR r -> row (lane/16)*8 + r, col = lane%16
__global__ void k_wmma_gemm_f32(const float* __restrict__ A,
                                const float* __restrict__ B,
                                float* __restrict__ C,
                                int M, int N, int K) {
  const int lane = threadIdx.x & 31;
  const int wave = threadIdx.x >> 5;
  const int wavesPerBlock = blockDim.x >> 5;
  const int tilesN = N >> 4;
  const int tile = blockIdx.x * wavesPerBlock + wave;
  if (tile >= (M >> 4) * tilesN) return;  // wave-uniform guard: EXEC stays all-1s

  const int tm = tile / tilesN;
  const int tn = tile - tm * tilesN;
  const int mn = lane & 15;            // row index within A frag / col within B frag
  const int kb = (lane >> 4) << 1;     // 0 for lanes 0-15, 2 for lanes 16-31

  const float* __restrict__ arow = A + (size_t)((tm << 4) + mn) * K;
  const float* __restrict__ bcol = B + (tn << 4) + mn;

  v8f acc = {};
  for (int k = 0; k < K; k += 4) {
    v2f a, b;
    a.x = arow[k + kb];
    a.y = arow[k + kb + 1];
    b.x = bcol[(size_t)(k + kb) * N];
    b.y = bcol[(size_t)(k + kb + 1) * N];
    acc = __builtin_amdgcn_wmma_f32_16x16x4_f32(
        /*neg_a=*/false, a, /*neg_b=*/false, b,
        /*c_mod=*/(short)0, acc, /*reuse_a=*/false, /*reuse_b=*/false);
  }

  const int row0 = (tm << 4) + ((lane >> 4) << 3);  // +0 or +8
  const int col  = (tn << 4) + mn;
#pragma unroll
  for (int r = 0; r < 8; ++r) {
    C[(size_t)(row0 + r) * N + col] = acc[r];
  }
}

// ------------------------- edge message scatter ----------------------------
// ACC[dst] += H[src] * dinv[src] * dinv[dst], one float4 chunk per thread.
// Consecutive threads cover consecutive chunks of the same edge -> the random
// source row is read contiguously; atomics go to L2 (192 MB keeps the full
// ACC matrix resident).
__global__ void k_scatter_edges(const float* __restrict__ H,
                                const int* __restrict__ src,
                                const int* __restrict__ dst,
                                const float* __restrict__ dinv,
                                float* __restrict__ ACC,
                                int E, int F) {
  const int chunks = F >> 2;
  long long gid = (long long)blockIdx.x * blockDim.x + threadIdx.x;
  if (gid >= (long long)E * chunks) return;
  const int e = (int)(gid / chunks);
  const int c = ((int)(gid - (long long)e * chunks)) << 2;
  const int s = src[e];
  const int d = dst[e];
  const float nrm = dinv[s] * dinv[d];
  const float4 h = *(const float4*)(H + (size_t)s * F + c);
  float* out = ACC + (size_t)d * F + c;
  unsafeAtomicAdd(out + 0, h.x * nrm);
  unsafeAtomicAdd(out + 1, h.y * nrm);
  unsafeAtomicAdd(out + 2, h.z * nrm);
  unsafeAtomicAdd(out + 3, h.w * nrm);
}

// ----------------------- self-loop + bias + ReLU ---------------------------
// OUT[i,c] = relu(ACC[i,c] + H[i,c]*dinv[i]^2 + bias[c]); safe in-place.
__global__ void k_epilogue(const float* __restrict__ ACC,
                           const float* __restrict__ H,
                           const float* __restrict__ dinv,
                           const float* __restrict__ bias,
                           float* __restrict__ OUT,
                           int Nn, int F) {
  long long gid = (long long)blockIdx.x * blockDim.x + threadIdx.x;
  if (gid >= (long long)Nn * F) return;
  const int i = (int)(gid / F);
  const int c = (int)(gid - (long long)i * F);
  const float di = dinv[i];
  const float v = ACC[gid] + H[gid] * (di * di) + bias[c];
  OUT[gid] = v > 0.0f ? v : 0.0f;
}

// ---------------------------------------------------------------------------
extern "C" void kernel_launch(void* const* d_in, const int* in_sizes, int n_in,
                              void* d_out, int out_size, void* d_ws, size_t ws_size,
                              hipStream_t stream) {
  (void)n_in; (void)out_size; (void)ws_size;

  const float* x  = (const float*)d_in[0];   // [N, 256]
  const int*   ei = (const int*)d_in[1];     // [2, E] (int32 per harness convention)
  const float* W1 = (const float*)d_in[2];   // [256, 256]
  const float* b1 = (const float*)d_in[3];   // [256]
  const float* W2 = (const float*)d_in[4];   // [256, 128]
  const float* b2 = (const float*)d_in[5];   // [128]

  const int N    = in_sizes[0] / 256;        // 50000
  const int E    = in_sizes[1] / 2;          // 800000
  const int DIN  = 256, DH = 256, DOUT = 128;
  const int* src = ei;
  const int* dst = ei + E;

  // Workspace carve-out (256B aligned slices).
  size_t off = 0;
  auto carve = [&](size_t bytes) {
    void* p = (char*)d_ws + off;
    off += (bytes + 255) & ~(size_t)255;
    return p;
  };
  float* dinv = (float*)carve((size_t)N * 4);          // degrees -> rsqrt in place
  float* H1   = (float*)carve((size_t)N * DH * 4);     // X @ W1
  float* ACC1 = (float*)carve((size_t)N * DH * 4);     // aggregated, then A1 in place
  float* H2   = (float*)carve((size_t)N * DOUT * 4);   // A1 @ W2
  float* ACC2 = (float*)carve((size_t)N * DOUT * 4);

  const int TB = 256;  // 8 waves per block (wave32)

  // Degrees / normalization.
  k_init_deg<<<(N + TB - 1) / TB, TB, 0, stream>>>(dinv, N);
  k_count_deg<<<(E + TB - 1) / TB, TB, 0, stream>>>(dst, dinv, E);
  k_deg_to_dinv<<<(N + TB - 1) / TB, TB, 0, stream>>>(dinv, N);

  // ---- Layer 1 ----
  hipMemsetAsync(ACC1, 0, (size_t)N * DH * 4, stream);
  {
    int tiles = (N >> 4) * (DH >> 4);  // 50000 tiles, 8 waves/block
    k_wmma_gemm_f32<<<(tiles + 7) / 8, TB, 0, stream>>>(x, W1, H1, N, DH, DIN);
  }
  {
    long long total = (long long)E * (DH >> 2);
    k_scatter_edges<<<(int)((total + TB - 1) / TB), TB, 0, stream>>>(
        H1, src, dst, dinv, ACC1, E, DH);
  }
  {
    long long total = (long long)N * DH;
    k_epilogue<<<(int)((total + TB - 1) / TB), TB, 0, stream>>>(
        ACC1, H1, dinv, b1, ACC1, N, DH);  // A1 in place
  }

  // ---- Layer 2 ----
  hipMemsetAsync(ACC2, 0, (size_t)N * DOUT * 4, stream);
  {
    int tiles = (N >> 4) * (DOUT >> 4);  // 25000 tiles
    k_wmma_gemm_f32<<<(tiles + 7) / 8, TB, 0, stream>>>(ACC1, W2, H2, N, DOUT, DH);
  }
  {
    long long total = (long long)E * (DOUT >> 2);
    k_scatter_edges<<<(int)((total + TB - 1) / TB), TB, 0, stream>>>(
        H2, src, dst, dinv, ACC2, E, DOUT);
  }
  {
    long long total = (long long)N * DOUT;
    k_epilogue<<<(int)((total + TB - 1) / TB), TB, 0, stream>>>(
        ACC2, H2, dinv, b2, (float*)d_out, N, DOUT);
  }
}